// NeurEPDiff3D_15247133901508
// MI455X (gfx1250) — compile-verified
//
#include <hip/hip_runtime.h>
#include <math.h>

#define NP    135168      // 64*64*33
#define NB    8
#define NPIX_TILES 8448   // NP/16
#define NMODE 2048        // 16*16*8 corner modes
#define NREAL 3584        // 16*16*14 real corner grid

typedef __attribute__((ext_vector_type(2))) float v2f;
typedef __attribute__((ext_vector_type(8))) float v8f;

static __device__ __forceinline__ v8f wmma4(v2f a, v2f b, v8f c) {
  // D(16x16,f32) += A(16x4,f32) * B(4x16,f32)
  return __builtin_amdgcn_wmma_f32_16x16x4_f32(false, a, false, b, (short)0, c, false, false);
}
static __device__ __forceinline__ float gelu_f(float v) {
  return 0.5f * v * (1.0f + erff(v * 0.70710678118654752440f));
}

// ---------------- fc0 : 3 -> 20 complex channel lift ----------------
__global__ __launch_bounds__(256) void k_fc0(const float* __restrict__ xre,
                                             const float* __restrict__ xim,
                                             const float* __restrict__ f0,
                                             float* __restrict__ XR,
                                             float* __restrict__ XI) {
  int idx = blockIdx.x * 256 + threadIdx.x;
  int b = idx / NP, p = idx - b * NP;
  float ir[3], ii[3];
#pragma unroll
  for (int i = 0; i < 3; ++i) {
    ir[i] = xre[(b * 3 + i) * NP + p];
    ii[i] = xim[(b * 3 + i) * NP + p];
  }
#pragma unroll
  for (int o = 0; o < 20; ++o) {
    float ar = 0.f, ai = 0.f;
#pragma unroll
    for (int i = 0; i < 3; ++i) {
      float wr = f0[2 * (i * 20 + o)], wi = f0[2 * (i * 20 + o) + 1];
      ar += ir[i] * wr - ii[i] * wi;
      ai += ir[i] * wi + ii[i] * wr;
    }
    XR[(b * 20 + o) * NP + p] = ar;
    XI[(b * 20 + o) * NP + p] = ai;
  }
}

// ------------- corner gather + irfftn (separable direct DFT) -------------
__global__ __launch_bounds__(256) void k_ifft(const float* __restrict__ XR,
                                              const float* __restrict__ XI,
                                              float* __restrict__ R1) {
  __shared__ float Ar[2048], Ai[2048], Br[2048], Bi[2048];
  __shared__ float c16[16], s16[16], c14[14], s14[14];
  int tid = threadIdx.x;
  int t = blockIdx.x;                       // b*20 + channel
  if (tid < 16) { float a = 0.39269908169872414f * (float)tid; c16[tid] = cosf(a); s16[tid] = sinf(a); }
  if (tid < 14) { float a = 0.44879895051282760f * (float)tid; c14[tid] = cosf(a); s14[tid] = sinf(a); }
  for (int m = tid; m < 2048; m += 256) {   // gather corner modes
    int a = m >> 7, bc = (m >> 3) & 15, z = m & 7;
    int gx = a + ((a < 8) ? 0 : 48);
    int gy = bc + ((bc < 8) ? 0 : 48);
    int p = (gx * 64 + gy) * 33 + z;
    Ar[m] = XR[t * NP + p];
    Ai[m] = XI[t * NP + p];
  }
  __syncthreads();
  for (int m = tid; m < 2048; m += 256) {   // ifft along x (e^{+i})
    int x = m >> 7, bc = (m >> 3) & 15, z = m & 7;
    float sr = 0.f, si = 0.f;
#pragma unroll
    for (int a = 0; a < 16; ++a) {
      int k = (a * x) & 15;
      float c = c16[k], s = s16[k];
      int ix = (a * 16 + bc) * 8 + z;
      sr += Ar[ix] * c - Ai[ix] * s;
      si += Ar[ix] * s + Ai[ix] * c;
    }
    Br[m] = sr; Bi[m] = si;
  }
  __syncthreads();
  for (int m = tid; m < 2048; m += 256) {   // ifft along y
    int x = m >> 7, y = (m >> 3) & 15, z = m & 7;
    float sr = 0.f, si = 0.f;
#pragma unroll
    for (int q = 0; q < 16; ++q) {
      int k = (q * y) & 15;
      float c = c16[k], s = s16[k];
      int ix = (x * 16 + q) * 8 + z;
      sr += Br[ix] * c - Bi[ix] * s;
      si += Br[ix] * s + Bi[ix] * c;
    }
    Ar[m] = sr; Ai[m] = si;
  }
  __syncthreads();
  for (int s = tid; s < NREAL; s += 256) {  // irfft along z: 8 bins -> 14
    int xy = s / 14, z = s - xy * 14;
    int base = xy * 8;
    float acc = Ar[base] + (((z & 1) == 0) ? Ar[base + 7] : -Ar[base + 7]);
#pragma unroll
    for (int k = 1; k < 7; ++k) {
      int kk = (k * z) % 14;
      acc += 2.f * (Ar[base + k] * c14[kk] - Ai[base + k] * s14[kk]);
    }
    R1[t * NREAL + s] = acc * (1.0f / 3584.0f);
  }
}

// ------------- per-position 20x20 real channel mix -------------
__global__ __launch_bounds__(256) void k_hmat(const float* __restrict__ R1,
                                              const float* __restrict__ hw,
                                              float* __restrict__ R2) {
  int idx = blockIdx.x * 256 + threadIdx.x;
  int b = idx / (20 * NREAL);
  int rem = idx - b * (20 * NREAL);
  int o = rem / NREAL;
  int s = rem - o * NREAL;
  float acc = 0.f;
#pragma unroll
  for (int i = 0; i < 20; ++i)
    acc += R1[(b * 20 + i) * NREAL + s] * hw[(i * 20 + o) * NREAL + s];
  R2[(b * 20 + o) * NREAL + s] = acc;
}

// ------------- rfftn (separable direct DFT) -------------
__global__ __launch_bounds__(256) void k_fft(const float* __restrict__ R2,
                                             float* __restrict__ HRe,
                                             float* __restrict__ HIm) {
  __shared__ float R[NREAL];
  __shared__ float Cr[2048], Ci[2048], Dr[2048], Di[2048];
  __shared__ float c16[16], s16[16], c14[14], s14[14];
  int tid = threadIdx.x;
  int t = blockIdx.x;
  if (tid < 16) { float a = 0.39269908169872414f * (float)tid; c16[tid] = cosf(a); s16[tid] = sinf(a); }
  if (tid < 14) { float a = 0.44879895051282760f * (float)tid; c14[tid] = cosf(a); s14[tid] = sinf(a); }
  for (int s = tid; s < NREAL; s += 256) R[s] = R2[t * NREAL + s];
  __syncthreads();
  for (int m = tid; m < 2048; m += 256) {   // rfft along z (e^{-i})
    int xy = m >> 3, kz = m & 7;
    float sr = 0.f, si = 0.f;
    int base = xy * 14;
#pragma unroll
    for (int z = 0; z < 14; ++z) {
      int kk = (kz * z) % 14;
      float v = R[base + z];
      sr += v * c14[kk];
      si -= v * s14[kk];
    }
    Cr[m] = sr; Ci[m] = si;
  }
  __syncthreads();
  for (int m = tid; m < 2048; m += 256) {   // fft along y (e^{-i})
    int x = m >> 7, ky = (m >> 3) & 15, kz = m & 7;
    float sr = 0.f, si = 0.f;
#pragma unroll
    for (int y = 0; y < 16; ++y) {
      int k = (y * ky) & 15;
      float c = c16[k], s = s16[k];
      int ix = (x * 16 + y) * 8 + kz;
      sr += Cr[ix] * c + Ci[ix] * s;
      si += Ci[ix] * c - Cr[ix] * s;
    }
    Dr[m] = sr; Di[m] = si;
  }
  __syncthreads();
  for (int m = tid; m < 2048; m += 256) {   // fft along x (e^{-i})
    int kx = m >> 7, ky = (m >> 3) & 15, kz = m & 7;
    float sr = 0.f, si = 0.f;
#pragma unroll
    for (int x = 0; x < 16; ++x) {
      int k = (x * kx) & 15;
      float c = c16[k], s = s16[k];
      int ix = (x * 16 + ky) * 8 + kz;
      sr += Dr[ix] * c + Di[ix] * s;
      si += Di[ix] * c - Dr[ix] * s;
    }
    HRe[t * NMODE + m] = sr;
    HIm[t * NMODE + m] = si;
  }
}

// ------------- layer: (H(x) + x*W) * S [+ gelu], in place, WMMA -------------
__global__ __launch_bounds__(256) void k_layer(float* __restrict__ XR,
                                               float* __restrict__ XI,
                                               const float* __restrict__ w,
                                               const float* __restrict__ HRe,
                                               const float* __restrict__ HIm,
                                               const float* __restrict__ sre,
                                               const float* __restrict__ sim,
                                               int do_gelu) {
  __shared__ float tb[8][2][20 * 17];
  int lane = threadIdx.x & 31;
  int wv = threadIdx.x >> 5;
  int tile = blockIdx.x * 8 + wv;
  int b = tile / NPIX_TILES;
  int p0 = (tile - b * NPIX_TILES) * 16;
  int col = lane & 15;
  int krow = (lane >> 4) << 1;   // K pair base: 0 or 2
  int mro = (lane >> 4) << 3;    // D-row offset: 0 or 8

  // A fragments: 16 pixels x 20 channels (re, im, -im)
  v2f are[5], aim[5], ain[5];
  {
    int p = p0 + col;
#pragma unroll
    for (int ks = 0; ks < 5; ++ks) {
      int c = ks * 4 + krow;
      int base = (b * 20 + c) * NP + p;
      v2f r; r.x = XR[base]; r.y = XR[base + NP];
      v2f i; i.x = XI[base]; i.y = XI[base + NP];
      are[ks] = r; aim[ks] = i; ain[ks] = -i;
    }
  }
  // hoist per-pixel smooth factor + corner-mode index
  float sr8[8], si8[8]; int mc[8];
#pragma unroll
  for (int r = 0; r < 8; ++r) {
    int p = p0 + r + mro;
    sr8[r] = sre[p]; si8[r] = sim[p];
    int z = p % 33; int rem = p / 33; int gy = rem & 63; int gx = rem >> 6;
    bool corner = (z < 8) && (gx < 8 || gx >= 56) && (gy < 8 || gy >= 56);
    int a = (gx < 8) ? gx : gx - 48;
    int bc = (gy < 8) ? gy : gy - 48;
    mc[r] = corner ? ((a * 16 + bc) * 8 + z) : -1;
  }

#pragma unroll
  for (int n = 0; n < 2; ++n) {
    int o = n * 16 + col;
    bool ov = (o < 20);
    v2f bre[5], bim[5];
#pragma unroll
    for (int ks = 0; ks < 5; ++ks) {
      int k = ks * 4 + krow;
      v2f br = {0.f, 0.f}, bi = {0.f, 0.f};
      if (ov) {
        br.x = w[2 * (k * 20 + o)];       br.y = w[2 * ((k + 1) * 20 + o)];
        bi.x = w[2 * (k * 20 + o) + 1];   bi.y = w[2 * ((k + 1) * 20 + o) + 1];
      }
      bre[ks] = br; bim[ks] = bi;
    }
    v8f cr = {}, ci = {};
#pragma unroll
    for (int ks = 0; ks < 5; ++ks) {
      cr = wmma4(are[ks], bre[ks], cr);
      cr = wmma4(ain[ks], bim[ks], cr);
      ci = wmma4(are[ks], bim[ks], ci);
      ci = wmma4(aim[ks], bre[ks], ci);
    }
#pragma unroll
    for (int r = 0; r < 8; ++r) {
      float tr = cr[r], ti = ci[r];
      if (ov && mc[r] >= 0) {
        tr += HRe[(b * 20 + o) * NMODE + mc[r]];
        ti += HIm[(b * 20 + o) * NMODE + mc[r]];
      }
      float ur = tr * sr8[r] - ti * si8[r];
      float ui = tr * si8[r] + ti * sr8[r];
      if (do_gelu) { ur = gelu_f(ur); ui = gelu_f(ui); }
      if (ov) {
        int pix = r + mro;
        tb[wv][0][o * 17 + pix] = ur;
        tb[wv][1][o * 17 + pix] = ui;
      }
    }
  }
  __syncthreads();
#pragma unroll
  for (int cc = 0; cc < 10; ++cc) {        // coalesced in-place store
    int c = cc * 2 + (lane >> 4);
    int pix = lane & 15;
    int g = (b * 20 + c) * NP + p0 + pix;
    XR[g] = tb[wv][0][c * 17 + pix];
    XI[g] = tb[wv][1][c * 17 + pix];
  }
}

// ------------- fused fc1 (20->128) + cgelu + fc2 (128->3), WMMA -------------
__global__ __launch_bounds__(256) void k_final(const float* __restrict__ XR,
                                               const float* __restrict__ XI,
                                               const float* __restrict__ f1,
                                               const float* __restrict__ f2,
                                               float* __restrict__ out) {
  __shared__ float yb[8][2][16 * 132];     // padded stride 132: conflict-free
  int lane = threadIdx.x & 31;
  int wv = threadIdx.x >> 5;
  int tile = blockIdx.x * 8 + wv;
  int b = tile / NPIX_TILES;
  int p0 = (tile - b * NPIX_TILES) * 16;
  int col = lane & 15;
  int krow = (lane >> 4) << 1;
  int mro = (lane >> 4) << 3;

  v2f are[5], aim[5], ain[5];
  {
    int p = p0 + col;
#pragma unroll
    for (int ks = 0; ks < 5; ++ks) {
      int c = ks * 4 + krow;
      int base = (b * 20 + c) * NP + p;
      v2f r; r.x = XR[base]; r.y = XR[base + NP];
      v2f i; i.x = XI[base]; i.y = XI[base + NP];
      are[ks] = r; aim[ks] = i; ain[ks] = -i;
    }
  }
  for (int n = 0; n < 8; ++n) {            // fc1: 8 N-tiles of 16
    int o = n * 16 + col;
    v2f bre[5], bim[5];
#pragma unroll
    for (int ks = 0; ks < 5; ++ks) {
      int k = ks * 4 + krow;
      v2f br, bi;
      br.x = f1[2 * (k * 128 + o)];     br.y = f1[2 * ((k + 1) * 128 + o)];
      bi.x = f1[2 * (k * 128 + o) + 1]; bi.y = f1[2 * ((k + 1) * 128 + o) + 1];
      bre[ks] = br; bim[ks] = bi;
    }
    v8f cr = {}, ci = {};
#pragma unroll
    for (int ks = 0; ks < 5; ++ks) {
      cr = wmma4(are[ks], bre[ks], cr);
      cr = wmma4(ain[ks], bim[ks], cr);
      ci = wmma4(are[ks], bim[ks], ci);
      ci = wmma4(aim[ks], bre[ks], ci);
    }
#pragma unroll
    for (int r = 0; r < 8; ++r) {
      int pix = r + mro;
      yb[wv][0][pix * 132 + o] = gelu_f(cr[r]);
      yb[wv][1][pix * 132 + o] = gelu_f(ci[r]);
    }
  }
  __syncthreads();
  v8f cr2 = {}, ci2 = {};                  // fc2: K=128 in 32 K4-steps
  bool ov = (col < 3);
  for (int ks = 0; ks < 32; ++ks) {
    int k = ks * 4 + krow;
    v2f a_r, a_i;
    a_r.x = yb[wv][0][col * 132 + k];  a_r.y = yb[wv][0][col * 132 + k + 1];
    a_i.x = yb[wv][1][col * 132 + k];  a_i.y = yb[wv][1][col * 132 + k + 1];
    v2f a_n = -a_i;
    v2f br = {0.f, 0.f}, bi = {0.f, 0.f};
    if (ov) {
      br.x = f2[2 * (k * 3 + col)];     br.y = f2[2 * ((k + 1) * 3 + col)];
      bi.x = f2[2 * (k * 3 + col) + 1]; bi.y = f2[2 * ((k + 1) * 3 + col) + 1];
    }
    cr2 = wmma4(a_r, br, cr2);
    cr2 = wmma4(a_n, bi, cr2);
    ci2 = wmma4(a_r, bi, ci2);
    ci2 = wmma4(a_i, br, ci2);
  }
  float2* o2 = (float2*)out;               // interleaved complex64
#pragma unroll
  for (int r = 0; r < 8; ++r) {
    if (ov) {
      int p = p0 + r + mro;
      o2[(b * 3 + col) * NP + p] = make_float2(cr2[r], ci2[r]);
    }
  }
}

extern "C" void kernel_launch(void* const* d_in, const int* in_sizes, int n_in,
                              void* d_out, int out_size, void* d_ws, size_t ws_size,
                              hipStream_t stream) {
  (void)in_sizes; (void)n_in; (void)out_size; (void)ws_size;
  const float* xre = (const float*)d_in[0];
  const float* xim = (const float*)d_in[1];
  const float* sre = (const float*)d_in[2];
  const float* sim = (const float*)d_in[3];
  const float* fc0 = (const float*)d_in[4];
  const float* wl[4]  = {(const float*)d_in[5], (const float*)d_in[6],
                         (const float*)d_in[7], (const float*)d_in[8]};
  const float* hwl[4] = {(const float*)d_in[9], (const float*)d_in[10],
                         (const float*)d_in[11], (const float*)d_in[12]};
  const float* fc1 = (const float*)d_in[13];
  const float* fc2 = (const float*)d_in[14];

  float* ws  = (float*)d_ws;
  float* XR  = ws;
  float* XI  = XR  + (size_t)NB * 20 * NP;
  float* R1  = XI  + (size_t)NB * 20 * NP;
  float* R2  = R1  + (size_t)NB * 20 * NREAL;
  float* HRe = R2  + (size_t)NB * 20 * NREAL;
  float* HIm = HRe + (size_t)NB * 20 * NMODE;

  k_fc0<<<NB * NP / 256, 256, 0, stream>>>(xre, xim, fc0, XR, XI);
  for (int l = 0; l < 4; ++l) {
    k_ifft<<<NB * 20, 256, 0, stream>>>(XR, XI, R1);
    k_hmat<<<NB * 20 * NREAL / 256, 256, 0, stream>>>(R1, hwl[l], R2);
    k_fft<<<NB * 20, 256, 0, stream>>>(R2, HRe, HIm);
    k_layer<<<NPIX_TILES, 256, 0, stream>>>(XR, XI, wl[l], HRe, HIm, sre, sim,
                                            (l < 3) ? 1 : 0);
  }
  k_final<<<NPIX_TILES, 256, 0, stream>>>(XR, XI, fc1, fc2, (float*)d_out);
}